// Head_386547056707
// MI455X (gfx1250) — compile-verified
//
#include <hip/hip_runtime.h>
#include <hip/hip_bf16.h>

// ---------------------------------------------------------------------------
// Single-head attention with ALiBi + causal mask (B=8, T=2048, C=512, H=64).
// Outputs: output [B,T,H] then attention [B,T,T], both fp32, concatenated.
//
// MI455X roofline: the mandatory attention-matrix write is 134 MB -> ~6us
// floor at 23.3 TB/s. Total matmul work is ~7.5 GFLOP (causal-clipped), so
// fp32 WMMA (v_wmma_f32_16x16x4_f32) keeps full reference precision while
// staying near the bandwidth bound. Scores live in LDS (16 x 2048 fp32 per
// block, ~132 KB of the 320 KB WGP LDS) through softmax and AV, so the
// attention matrix is written to HBM exactly once, fully coalesced.
// ---------------------------------------------------------------------------

typedef float v2f __attribute__((ext_vector_type(2)));
typedef float v8f __attribute__((ext_vector_type(8)));

#define T_SEQ   2048
#define C_EMB   512
#define H_DIM   64
#define B_BAT   8
#define QSZ     (B_BAT * T_SEQ * H_DIM)       // elements per q/k/v buffer
#define SPITCH  2052                          // 2048+4: AV column reads hit distinct banks
#define SCALE_C 0.044194173824159216f         // 1/sqrt(512)

static __device__ __forceinline__ v2f ld2(const float* p) {
    return *(const v2f*)p;
}

static __device__ __forceinline__ v8f wmma_f32(v2f a, v2f b, v8f c) {
    // D(16x16) = A(16x4) x B(4x16) + C, all fp32.
    return __builtin_amdgcn_wmma_f32_16x16x4_f32(
        false, a, false, b, (short)0, c, false, false);
}

// ---------------------------------------------------------------------------
// Kernel 1: q/k/v = x @ W{q,k,v}.  M=16384, K=512, N=64.
// One wave computes a full 16x64 row-tile of one matrix (4 accumulators),
// so each x A-fragment is loaded once and reused across the 4 H-tiles.
// 3 mats * 1024 row-tiles = 3072 waves; 8 waves/block -> 384 blocks.
// ---------------------------------------------------------------------------
__global__ __launch_bounds__(256) void qkv_proj_kernel(
    const float* __restrict__ x,
    const float* __restrict__ Wq,
    const float* __restrict__ Wk,
    const float* __restrict__ Wv,
    float* __restrict__ qkv)
{
    const int tid  = threadIdx.x;
    const int wv   = tid >> 5;
    const int lane = tid & 31;
    const int lo   = lane & 15;     // row/col within half-wave
    const int hi   = lane >> 4;     // half-wave select (K pair)

    const int gw    = blockIdx.x * 8 + wv;   // [0, 3072)
    const int mat   = gw >> 10;              // 0=q,1=k,2=v
    const int m0    = (gw & 1023) << 4;      // row-tile * 16

    const float* W = (mat == 0) ? Wq : ((mat == 1) ? Wk : Wv);
    float* out = qkv + (size_t)mat * QSZ;

    v8f acc0 = {}, acc1 = {}, acc2 = {}, acc3 = {};
    const float* xrow = x + (size_t)(m0 + lo) * C_EMB + 2 * hi;
    const float* wrow = W + (size_t)(2 * hi) * H_DIM + lo;

    for (int kk = 0; kk < C_EMB; kk += 4) {
        // A fragment: lane holds x[m0+lo][kk + 2*hi .. +1]
        v2f a = ld2(xrow + kk);
        // B fragments for all 4 H-tiles: W[kk+2*hi][n0+lo], W[kk+2*hi+1][n0+lo]
        const float* wp = wrow + (size_t)kk * H_DIM;
        v2f b0 = { wp[0],  wp[H_DIM] };
        v2f b1 = { wp[16], wp[16 + H_DIM] };
        v2f b2 = { wp[32], wp[32 + H_DIM] };
        v2f b3 = { wp[48], wp[48 + H_DIM] };
        acc0 = wmma_f32(a, b0, acc0);
        acc1 = wmma_f32(a, b1, acc1);
        acc2 = wmma_f32(a, b2, acc2);
        acc3 = wmma_f32(a, b3, acc3);
    }
    // C/D layout: VGPR r -> (M = r + 8*hi, N = lo)
    for (int r = 0; r < 8; ++r) {
        const size_t row = (size_t)(m0 + r + 8 * hi) * H_DIM + lo;
        out[row +  0] = acc0[r];
        out[row + 16] = acc1[r];
        out[row + 32] = acc2[r];
        out[row + 48] = acc3[r];
    }
}

// ---------------------------------------------------------------------------
// Kernel 2: per (batch, 16-row query block): scores -> softmax -> single
// coalesced attention write -> output = attn @ v. Scores resident in LDS.
// grid = (128, 8), block = 256 (8 waves). Causal work is interleaved across
// waves and all scans are clipped to kend = i0+16.
// ---------------------------------------------------------------------------
__global__ __launch_bounds__(256) void attn_kernel(
    const float* __restrict__ qg,
    const float* __restrict__ kg,
    const float* __restrict__ vg,
    float* __restrict__ outp,
    float* __restrict__ att)
{
    __shared__ float S[16 * SPITCH];    // exp-scores (cols [0,kend) valid)
    __shared__ float qs[16 * H_DIM];    // query block
    __shared__ float P[16 * H_DIM];     // split-K partial for AV
    __shared__ float rowinv[16];        // 1/rowsum

    const int b    = blockIdx.y;
    const int i0   = blockIdx.x * 16;
    const int kend = i0 + 16;           // causal extent of this row block

    const float* q = qg + ((size_t)b * T_SEQ + i0) * H_DIM;
    const float* k = kg + (size_t)b * T_SEQ * H_DIM;
    const float* v = vg + (size_t)b * T_SEQ * H_DIM;

    const int tid  = threadIdx.x;
    const int wv   = tid >> 5;
    const int lane = tid & 31;
    const int lo   = lane & 15;
    const int hi   = lane >> 4;

    // Stage the 16x64 query block in LDS.
    for (int idx = tid; idx < 16 * H_DIM; idx += 256) qs[idx] = q[idx];
    __syncthreads();

    // Preload all A fragments of the q block (K=64 -> 16 steps of K=4).
    v2f qa[16];
    for (int t = 0; t < 16; ++t)
        qa[t] = ld2(&qs[lo * H_DIM + t * 4 + 2 * hi]);

    // ---- scores: S[m][j] = (q.k + (j - row)) * scale, per-element masked.
    // Tiles interleaved across waves (jt = wv, wv+8, ...) for load balance.
    for (int jt = wv; jt <= blockIdx.x; jt += 8) {
        const int j0 = jt * 16;
        if (jt + 8 <= blockIdx.x)   // prefetch next tile's k rows (L2 hint)
            __builtin_prefetch(&k[(size_t)(j0 + 128 + lo) * H_DIM + 2 * hi], 0, 1);
        v8f acc = {};
        for (int t = 0; t < 16; ++t) {
            // B = k^T tile: lane holds k[j0+lo][4t + 2*hi .. +1]
            v2f bk = ld2(&k[(size_t)(j0 + lo) * H_DIM + t * 4 + 2 * hi]);
            acc = wmma_f32(qa[t], bk, acc);
        }
        for (int r = 0; r < 8; ++r) {
            const int m   = r + 8 * hi;
            const int row = i0 + m;
            const int col = j0 + lo;
            float val = (col <= row)
                        ? (acc[r] + (float)(col - row)) * SCALE_C
                        : -1e30f;       // exp -> 0
            S[m * SPITCH + col] = val;
        }
    }
    __syncthreads();

    // ---- row softmax over [0, kend): wave wv owns rows 2*wv + hi ----
    {
        const int r = 2 * wv + hi;
        float mx = -1e30f;
        for (int s = lo; s < kend; s += 16)
            mx = fmaxf(mx, S[r * SPITCH + s]);
        for (int m = 1; m < 16; m <<= 1)
            mx = fmaxf(mx, __shfl_xor(mx, m, 16));
        float sum = 0.0f;
        for (int s = lo; s < kend; s += 16) {
            float e = __expf(S[r * SPITCH + s] - mx);
            S[r * SPITCH + s] = e;                  // keep unnormalized exp
            sum += e;
        }
        for (int m = 1; m < 16; m <<= 1)
            sum += __shfl_xor(sum, m, 16);
        if (lo == 0) rowinv[r] = 1.0f / sum;
    }
    __syncthreads();

    // ---- the big write: normalized attention, fully coalesced; zeros past
    // the causal extent come from literals, not LDS.
    {
        float* arow = att + ((size_t)b * T_SEQ + i0) * T_SEQ;
        for (int idx = tid; idx < 16 * T_SEQ; idx += 256) {
            const int m = idx >> 11;
            const int n = idx & (T_SEQ - 1);
            float val = (n < kend) ? S[m * SPITCH + n] * rowinv[m] : 0.0f;
            arow[(size_t)m * T_SEQ + n] = val;
        }
    }

    // ---- output = (e @ v) * rowinv.  Split-K at kend/2: equal work for
    // both wave groups regardless of block position.
    const int nt    = wv & 3;                 // H tile (4 x 16)
    const int kh    = wv >> 2;                // K half
    const int h0    = nt * 16;
    const int khalf = kend >> 1;              // multiple of 8
    const int kbeg  = kh * khalf;
    const int klim  = kbeg + khalf;

    v8f acc = {};
    for (int kk = kbeg; kk < klim; kk += 4) {
        if (kk + 64 < klim)                   // stream v ahead of the WMMAs
            __builtin_prefetch(&v[(size_t)(kk + 64 + 2 * hi) * H_DIM + h0 + lo], 0, 1);
        // A = e tile: lane holds S[lo][kk + 2*hi .. +1]
        v2f ae = ld2(&S[lo * SPITCH + kk + 2 * hi]);
        // B = v tile: lane holds v[kk+2*hi][h0+lo], v[kk+2*hi+1][h0+lo]
        float b0 = v[(size_t)(kk + 2 * hi) * H_DIM + h0 + lo];
        float b1 = v[(size_t)(kk + 2 * hi + 1) * H_DIM + h0 + lo];
        v2f bv = {b0, b1};
        acc = wmma_f32(ae, bv, acc);
    }
    if (kh == 1) {
        for (int r = 0; r < 8; ++r)
            P[(r + 8 * hi) * H_DIM + h0 + lo] = acc[r];
    }
    __syncthreads();
    if (kh == 0) {
        float* orow = outp + ((size_t)b * T_SEQ + i0) * H_DIM;
        for (int r = 0; r < 8; ++r) {
            const int m = r + 8 * hi;
            float val = (acc[r] + P[m * H_DIM + h0 + lo]) * rowinv[m];
            orow[(size_t)m * H_DIM + h0 + lo] = val;
        }
    }
}

// ---------------------------------------------------------------------------
extern "C" void kernel_launch(void* const* d_in, const int* in_sizes, int n_in,
                              void* d_out, int out_size, void* d_ws, size_t ws_size,
                              hipStream_t stream) {
    const float* x  = (const float*)d_in[0];
    const float* Wq = (const float*)d_in[1];
    const float* Wk = (const float*)d_in[2];
    const float* Wv = (const float*)d_in[3];

    float* ws = (float*)d_ws;
    float* qb = ws;
    float* kb = ws + QSZ;
    float* vb = ws + 2 * QSZ;

    float* outp = (float*)d_out;                          // [B,T,H]
    float* att  = outp + (size_t)B_BAT * T_SEQ * H_DIM;   // [B,T,T]

    // Phase 1: QKV projections. 3072 wave-tiles / 8 waves per block.
    qkv_proj_kernel<<<384, 256, 0, stream>>>(x, Wq, Wk, Wv, ws);

    // Phase 2: scores + softmax + attention write + AV.
    attn_kernel<<<dim3(T_SEQ / 16, B_BAT), 256, 0, stream>>>(qb, kb, vb, outp, att);
}